// Guided_Conv_42176578847221
// MI455X (gfx1250) — compile-verified
//
#include <hip/hip_runtime.h>
#include <hip/hip_bf16.h>

typedef __attribute__((ext_vector_type(2))) float v2f;
typedef __attribute__((ext_vector_type(8))) float v8f;

#define FCH 9
#define PSZ 24
#define NPIX (PSZ * PSZ)        // 576
#define PATCH (NPIX * FCH)      // 5184 floats per patch (contiguous in in/out layout)
#define NV4   (PATCH / 4)       // 1296 float4s per patch
#define DCPAD 12                // depthwise output channels padded 9 -> 12 for WMMA K
#define DCSZ  (NPIX * DCPAD)    // 6912
#define NTHREADS 288            // 9 waves of 32

// One workgroup per patch (4096 patches). Fully fused, single HBM pass:
//  b128-stage guidance+depth -> dynamic kernel gen (conv + dense + L2 clips)
//  -> depthwise 3x3 (K-padded to 12 in LDS) -> 1x1 channel matmul via
//  chained V_WMMA_F32_16X16X4_F32 -> LDS re-pack -> coalesced b128 stores.
__global__ __launch_bounds__(NTHREADS)
void guided_conv_fused(const float* __restrict__ gin,
                       const float* __restrict__ din,
                       const float* __restrict__ conv_w,   // (3,3,9,9)
                       const float* __restrict__ conv_b,   // (9,)
                       const float* __restrict__ dense_w,  // (9,81)
                       const float* __restrict__ dense_b,  // (81,)
                       float* __restrict__ out)
{
    __shared__ __align__(16) float g[DCSZ];     // [0,PATCH): guidance; later dc[pix][12]
    __shared__ __align__(16) float dpt[PATCH];  // depth patch; later WMMA result staging
    __shared__ float cw[729];                   // conv_w staged
    __shared__ float dwm[729];                  // dense_w staged
    __shared__ float ck[81];                    // conv out -> normalized dw kernels [f*9 + dy*3+dx]
    __shared__ float part[NTHREADS];
    __shared__ float gap[FCH];
    __shared__ float dd[81];                    // dense out
    __shared__ float Mp[12 * 16];               // K-,N-padded channel-mix matrix (WMMA B)
    __shared__ float sM;                        // dense clip scale

    const int tid = threadIdx.x;
    const int n  = blockIdx.x;                  // flat patch id == output patch-major id
    const int b  = n >> 8;
    const int r  = n & 255;
    const int ih = r >> 4;
    const int iw = r & 15;
    const long base_in  = (((long)b * 384 + ih * PSZ) * 384 + iw * PSZ) * FCH;
    const long base_out = (long)n * PATCH;

    // ---- Stage 0: wide (b128) staging of both patches + small weights ----
    // Patch row = 216 contiguous floats, row starts are 16B-aligned; LDS idx == i*4.
    for (int i = tid; i < NV4; i += NTHREADS) {
        const int py = i / 54;                  // 54 float4 per row
        const int q  = i - py * 54;
        const long off = base_in + (long)py * (384 * FCH) + q * 4;
        const float4 gv = *reinterpret_cast<const float4*>(gin + off);
        const float4 dv = *reinterpret_cast<const float4*>(din + off);
        *reinterpret_cast<float4*>(g   + i * 4) = gv;
        *reinterpret_cast<float4*>(dpt + i * 4) = dv;
    }
    for (int i = tid; i < 729; i += NTHREADS) {
        cw[i]  = conv_w[i];
        dwm[i] = dense_w[i];
    }
    __syncthreads();

    // ---- Region A: stride-8 SAME conv (pad==0 for 24/8/3) + gap partial sums ----
    if (tid < 81) {
        const int oc = tid % 9;
        const int kx = (tid / 9) % 3;
        const int ky = tid / 27;
        float acc = conv_b[oc];
        #pragma unroll
        for (int dy = 0; dy < 3; ++dy)
            #pragma unroll
            for (int dx = 0; dx < 3; ++dx)
                for (int ic = 0; ic < FCH; ++ic)
                    acc += g[((ky * 8 + dy) * PSZ + (kx * 8 + dx)) * FCH + ic]
                         * cw[((dy * 3 + dx) * 9 + ic) * 9 + oc];
        ck[oc * 9 + ky * 3 + kx] = acc;         // spatial taps grouped per channel for the norm
    }
    {
        const int c = tid % 9;                  // 288 = 9 * 32 exactly
        const int s = tid / 9;                  // 0..31
        float acc = 0.f;
        for (int p = s; p < NPIX; p += 32) acc += g[p * FCH + c];
        part[tid] = acc;
    }
    __syncthreads();

    // ---- Region B: per-channel kernel L2 clip + gap reduce ----
    if (tid < 9) {
        float acc = 0.f;
        for (int s = 0; s < 32; ++s) acc += part[s * 9 + tid];
        gap[tid] = acc * (1.0f / (float)NPIX);
        float ss = 0.f;
        #pragma unroll
        for (int j = 0; j < 9; ++j) { float v = ck[tid * 9 + j]; ss += v * v; }
        const float scale = 1.0f / fmaxf(sqrtf(ss), 1.0f);   // tf.clip_by_norm(.., 1)
        #pragma unroll
        for (int j = 0; j < 9; ++j) ck[tid * 9 + j] *= scale;
    }
    __syncthreads();

    // ---- Region C: dense layer + depthwise 3x3 SAME -> dc over g, K-padded to 12 ----
    if (tid < 81) {
        float acc = dense_b[tid];
        #pragma unroll
        for (int i = 0; i < FCH; ++i) acc += gap[i] * dwm[i * 81 + tid];
        dd[tid] = acc;
    }
    for (int i = tid; i < DCSZ; i += NTHREADS) {     // 6912 = 288 * 24
        const int p  = i / DCPAD;
        const int f  = i - p * DCPAD;
        float acc = 0.f;
        if (f < FCH) {
            const int py = p / PSZ, px = p - py * PSZ;
            #pragma unroll
            for (int dy = 0; dy < 3; ++dy) {
                const int yy = py + dy - 1;
                if (yy < 0 || yy >= PSZ) continue;
                #pragma unroll
                for (int dx = 0; dx < 3; ++dx) {
                    const int xx = px + dx - 1;
                    if (xx < 0 || xx >= PSZ) continue;
                    acc += dpt[(yy * PSZ + xx) * FCH + f] * ck[f * 9 + dy * 3 + dx];
                }
            }
        }
        // NOTE: writes hit g[p*12+f] (p ascending through old guidance region);
        // g (guidance) reads all completed before this barrier region. pad ch -> 0.
        g[i] = acc;
    }
    __syncthreads();

    // ---- Region D: global L2 clip of dense output ----
    if (tid == 0) {
        float ss = 0.f;
        for (int j = 0; j < 81; ++j) ss += dd[j] * dd[j];
        sM = 1.0f / fmaxf(sqrtf(ss), 1.0f);
    }
    __syncthreads();

    // ---- Region E: build padded B matrix Mp[12][16] (K pad 9->12, N pad 9->16) ----
    if (tid < 192) {
        const int k = tid >> 4, o = tid & 15;
        Mp[tid] = (k < 9 && o < 9) ? dd[k * 9 + o] * sM : 0.0f;
    }
    __syncthreads();

    // ---- Region F: 1x1 cross-channel conv via chained V_WMMA_F32_16X16X4_F32 ----
    // 576 pixels -> 36 tiles of 16 rows; 9 waves x 4 tiles. K = 12 as 3 x K=4,
    // all fragment loads unconditional thanks to the 12-channel padding.
    const int wave = tid >> 5;            // 0..8
    const int lane = tid & 31;
    const int row  = lane & 15;           // A: M-row / B & D: N-col
    const int kk   = (lane >> 4) << 1;    // A/B K phase: 0 or 2
    const int mhi  = (lane >> 4) << 3;    // D: +8 row offset for upper half-wave

    for (int t = 0; t < 4; ++t) {
        const int tile = wave * 4 + t;
        const int pix0 = tile * 16;
        v8f acc = {0.f, 0.f, 0.f, 0.f, 0.f, 0.f, 0.f, 0.f};
        #pragma unroll
        for (int c = 0; c < 3; ++c) {
            v2f a, bm;
            #pragma unroll
            for (int j = 0; j < 2; ++j) {
                const int ch = c * 4 + kk + j;
                a[j]  = g[(pix0 + row) * DCPAD + ch];    // dc, zero-padded K
                bm[j] = Mp[ch * 16 + row];               // Mp, zero-padded K and N
            }
            acc = __builtin_amdgcn_wmma_f32_16x16x4_f32(
                /*neg_a=*/false, a, /*neg_b=*/false, bm,
                /*c_mod=*/(short)0, acc, /*reuse_a=*/false, /*reuse_b=*/false);
        }
        // Scatter D into LDS (depth buffer is dead) for coalesced wide stores.
        #pragma unroll
        for (int rr = 0; rr < 8; ++rr) {
            const int pix = pix0 + rr + mhi;  // D VGPR rr: lanes0-15 -> M=rr, lanes16-31 -> M=rr+8
            if (row < FCH)
                dpt[pix * FCH + row] = acc[rr];
        }
    }
    __syncthreads();

    // ---- Stage Z: coalesced b128 writeback of the whole patch ----
    for (int i = tid; i < NV4; i += NTHREADS) {
        *reinterpret_cast<float4*>(out + base_out + i * 4) =
            *reinterpret_cast<const float4*>(dpt + i * 4);
    }
}

extern "C" void kernel_launch(void* const* d_in, const int* in_sizes, int n_in,
                              void* d_out, int out_size, void* d_ws, size_t ws_size,
                              hipStream_t stream) {
    const float* guidance = (const float*)d_in[0];
    const float* depth    = (const float*)d_in[1];
    const float* conv_w   = (const float*)d_in[2];
    const float* conv_b   = (const float*)d_in[3];
    const float* dense_w  = (const float*)d_in[4];
    const float* dense_b  = (const float*)d_in[5];
    float* out = (float*)d_out;

    const int nblocks = 16 * 16 * 16;   // B * nh * nw = 4096 patches
    guided_conv_fused<<<nblocks, NTHREADS, 0, stream>>>(
        guidance, depth, conv_w, conv_b, dense_w, dense_b, out);
}